// NetGIN_17188459118904
// MI455X (gfx1250) — compile-verified
//
#include <hip/hip_runtime.h>
#include <hip/hip_bf16.h>
#include <math.h>

#define N_NODES  100000
#define N_EDGES  3200000
#define N_GRAPHS 512
#define N_FEAT   128
#define DIM      10
#define OUTD     16
#define PADD     16   // padded feature stride (pad cols kept at exactly 0)

typedef float v2f __attribute__((ext_vector_type(2)));
typedef float v8f __attribute__((ext_vector_type(8)));

// ---------------------------------------------------------------------------
// Zero fill
// ---------------------------------------------------------------------------
__global__ void fill_zero_kernel(float* __restrict__ p, int n) {
  int i = blockIdx.x * blockDim.x + threadIdx.x;
  if (i < n) p[i] = 0.0f;
}

// ---------------------------------------------------------------------------
// Per-graph node counts (float, for the mean-pool divide)
// ---------------------------------------------------------------------------
__global__ void count_kernel(const int* __restrict__ batch, float* __restrict__ cnt) {
  int i = blockIdx.x * blockDim.x + threadIdx.x;
  if (i < N_NODES) atomicAdd(&cnt[batch[i]], 1.0f);
}

// ---------------------------------------------------------------------------
// C[tiles*16, PADD] = A[tiles*16, LDA](:,0:WK) @ W[WK, wN]  (+bias, opt. relu)
// One wave per 16-row M tile; V_WMMA_F32_16X16X4_F32, K unrolled at compile
// time. B is staged once per block into LDS, transposed (Bt[n][k]) and
// zero-padded to KP, so the inner loop is branch-free: one global b64 load
// (A) + one ds b64 load (B) per WMMA. wN <= 16; output cols >= wN get exact 0.
// A rows must have >= KP readable floats (internal buffers are zero-padded).
// ---------------------------------------------------------------------------
template<int WK, int LDA>
__global__ __launch_bounds__(128) void gemm_wmma_f32(
    const float* __restrict__ A, const float* __restrict__ W, int wN,
    const float* __restrict__ bias, float* __restrict__ C,
    int nTiles, int relu) {
  constexpr int KP = ((WK + 3) / 4) * 4;    // padded K (multiple of 4)
  __shared__ float Bt[16 * KP];             // Bt[n][k], transposed + padded

  // Cooperative, branch-light staging of B (all waves, before any return)
  for (int idx = threadIdx.x; idx < 16 * KP; idx += 128) {
    int n = idx / KP;
    int k = idx % KP;
    Bt[idx] = (k < WK && n < wN) ? W[k * wN + n] : 0.f;
  }
  __syncthreads();

  int wave = (int)((blockIdx.x * 128u + threadIdx.x) >> 5);
  if (wave >= nTiles) return;               // wave-uniform: EXEC stays all-1s
  int lane = threadIdx.x & 31;
  int half = lane >> 4;                     // 0: K pair {0,1}, 1: {2,3}
  int mrow = lane & 15;                     // A row in tile / output column

  const float* Arow = A + (size_t)(wave * 16 + mrow) * (size_t)LDA + half * 2;
  const float* Brow = &Bt[mrow * KP + half * 2];

  v8f c = {0.f, 0.f, 0.f, 0.f, 0.f, 0.f, 0.f, 0.f};
  #pragma unroll
  for (int ks = 0; ks < KP / 4; ++ks) {
    // A 16x4 layout: lane m (+16 => k+2), vgpr v -> A[m][4ks + 2*half + v]
    v2f a = *(const v2f*)(Arow + ks * 4);   // aligned global_load_b64
    // B 4x16 layout (mirror): lane n (+16 => k+2) -> B[4ks + 2*half + v][n]
    v2f b = *(const v2f*)(Brow + ks * 4);   // aligned ds_load_b64
    c = __builtin_amdgcn_wmma_f32_16x16x4_f32(false, a, false, b, (short)0, c,
                                              false, false);
  }

  float bv = (bias != nullptr && mrow < wN) ? bias[mrow] : 0.f;
  int rbase = wave * 16 + half * 8;         // C/D layout: M = v + 8*(lane>>4)
  #pragma unroll
  for (int v = 0; v < 8; ++v) {
    float val = c[v] + bv;                  // pad cols: 0 + 0
    if (relu) val = val > 0.f ? val : 0.f;
    C[(size_t)(rbase + v) * PADD + mrow] = val;
  }
}

// ---------------------------------------------------------------------------
// agg[dst[e]][d] += y[src[e]][d]   one thread per (edge, d in 0..15), d<10 live
// ---------------------------------------------------------------------------
__global__ void edge_scatter_kernel(const int* __restrict__ src,
                                    const int* __restrict__ dst,
                                    const float* __restrict__ y,
                                    float* __restrict__ agg) {
  int i = blockIdx.x * blockDim.x + threadIdx.x;
  int e = i >> 4;
  int d = i & 15;
  if (e < N_EDGES && d < DIM) {
    atomicAdd(&agg[(size_t)dst[e] * PADD + d], y[(size_t)src[e] * PADD + d]);
  }
}

// ---------------------------------------------------------------------------
// y = relu(y + agg + b1)   (in place; pad cols stay 0)
// ---------------------------------------------------------------------------
__global__ void add_bias_relu_kernel(float* __restrict__ y,
                                     const float* __restrict__ agg,
                                     const float* __restrict__ b1) {
  int i = blockIdx.x * blockDim.x + threadIdx.x;
  if (i >= N_NODES * PADD) return;
  int d = i & 15;
  float bv = (d < DIM) ? b1[d] : 0.f;
  float v = y[i] + agg[i] + bv;
  y[i] = v > 0.f ? v : 0.f;
}

// ---------------------------------------------------------------------------
// pool[batch[node]][d] += h[node][d]
// ---------------------------------------------------------------------------
__global__ void pool_accum_kernel(const float* __restrict__ h,
                                  const int* __restrict__ batch,
                                  float* __restrict__ pool) {
  int i = blockIdx.x * blockDim.x + threadIdx.x;
  if (i >= N_NODES * PADD) return;
  int node = i >> 4;
  int d = i & 15;
  if (d < DIM) atomicAdd(&pool[(size_t)batch[node] * PADD + d], h[i]);
}

// ---------------------------------------------------------------------------
// out[g][o] = tanh( sum_l (pool_l[g]/cnt[g]) . L_l[:,o] )
// ---------------------------------------------------------------------------
__global__ void final_kernel(const float* __restrict__ pool,
                             const float* __restrict__ cnt,
                             const float* __restrict__ L1,
                             const float* __restrict__ L2,
                             const float* __restrict__ L3,
                             const float* __restrict__ L4,
                             const float* __restrict__ L5,
                             float* __restrict__ out) {
  int i = blockIdx.x * blockDim.x + threadIdx.x;
  if (i >= N_GRAPHS * OUTD) return;
  int g = i >> 4;
  int o = i & 15;
  float cg = cnt[g];
  cg = cg > 1.f ? cg : 1.f;
  const float* Ls[5] = {L1, L2, L3, L4, L5};
  float acc = 0.f;
  #pragma unroll
  for (int l = 0; l < 5; ++l) {
    const float* P = pool + (size_t)l * N_GRAPHS * PADD + (size_t)g * PADD;
    const float* L = Ls[l];
    #pragma unroll
    for (int d = 0; d < DIM; ++d) acc += (P[d] / cg) * L[d * OUTD + o];
  }
  out[i] = tanhf(acc);
}

// ---------------------------------------------------------------------------
extern "C" void kernel_launch(void* const* d_in, const int* in_sizes, int n_in,
                              void* d_out, int out_size, void* d_ws, size_t ws_size,
                              hipStream_t stream) {
  (void)in_sizes; (void)n_in; (void)out_size; (void)ws_size;

  const float* x     = (const float*)d_in[0];
  const int*   ei    = (const int*)d_in[1];
  const int*   src   = ei;             // edge_index[0]
  const int*   dst   = ei + N_EDGES;   // edge_index[1]
  const int*   batch = (const int*)d_in[2];

  const float *w1[5], *b1[5], *w2[5], *b2[5], *L[5];
  for (int l = 0; l < 5; ++l) {
    w1[l] = (const float*)d_in[3 + 4 * l];
    b1[l] = (const float*)d_in[4 + 4 * l];
    w2[l] = (const float*)d_in[5 + 4 * l];
    b2[l] = (const float*)d_in[6 + 4 * l];
    L[l]  = (const float*)d_in[23 + l];
  }

  // Workspace layout (floats): h | y | agg | pool[5][G][16] | cnt[G]
  float* ws   = (float*)d_ws;
  float* h    = ws;
  float* y    = h   + (size_t)N_NODES * PADD;
  float* agg  = y   + (size_t)N_NODES * PADD;
  float* pool = agg + (size_t)N_NODES * PADD;
  float* cnt  = pool + (size_t)5 * N_GRAPHS * PADD;

  const int T = 256;
  const int nTiles   = N_NODES / 16;                 // 6250 (exact)
  const int nElem    = N_NODES * PADD;               // 1.6M
  const int ewBlocks = (nElem + T - 1) / T;
  const int esBlocks = (N_EDGES * 16 + T - 1) / T;   // 200000
  dim3 gemmGrid((nTiles + 3) / 4), gemmBlock(128);   // 4 waves/block

  // pool + cnt are contiguous: zero both in one shot
  int nPoolCnt = 5 * N_GRAPHS * PADD + N_GRAPHS;
  fill_zero_kernel<<<(nPoolCnt + T - 1) / T, T, 0, stream>>>(pool, nPoolCnt);
  count_kernel<<<(N_NODES + T - 1) / T, T, 0, stream>>>(batch, cnt);

  for (int l = 0; l < 5; ++l) {
    // y = Ain @ w1  (linear-before-aggregate: segment_sum commutes with W1)
    if (l == 0) {
      gemm_wmma_f32<N_FEAT, N_FEAT><<<gemmGrid, gemmBlock, 0, stream>>>(
          x, w1[l], DIM, nullptr, y, nTiles, 0);
    } else {
      gemm_wmma_f32<DIM, PADD><<<gemmGrid, gemmBlock, 0, stream>>>(
          h, w1[l], DIM, nullptr, y, nTiles, 0);
    }
    fill_zero_kernel<<<ewBlocks, T, 0, stream>>>(agg, nElem);
    edge_scatter_kernel<<<esBlocks, T, 0, stream>>>(src, dst, y, agg);
    add_bias_relu_kernel<<<ewBlocks, T, 0, stream>>>(y, agg, b1[l]);
    // h = relu(y @ w2 + b2)
    gemm_wmma_f32<DIM, PADD><<<gemmGrid, gemmBlock, 0, stream>>>(
        y, w2[l], DIM, b2[l], h, nTiles, 1);
    pool_accum_kernel<<<ewBlocks, T, 0, stream>>>(
        h, batch, pool + (size_t)l * N_GRAPHS * PADD);
  }

  final_kernel<<<(N_GRAPHS * OUTD + T - 1) / T, T, 0, stream>>>(
      pool, cnt, L[0], L[1], L[2], L[3], L[4], (float*)d_out);
}